// RGCNAggregator_47777216201142
// MI455X (gfx1250) — compile-verified
//
#include <hip/hip_runtime.h>
#include <math.h>

// ---------------------------------------------------------------------------
// Problem constants (match reference)
// ---------------------------------------------------------------------------
#define H_DIM     128
#define RRELU_SLP 0.2291666666666667f   // 11/48

typedef __attribute__((ext_vector_type(2))) float v2f;
typedef __attribute__((ext_vector_type(8))) float v8f;

// ---------------------------------------------------------------------------
// Zero a float buffer (grid-stride)
// ---------------------------------------------------------------------------
__global__ void zero_f32(float* __restrict__ p, size_t n) {
    size_t i = (size_t)blockIdx.x * blockDim.x + threadIdx.x;
    size_t stride = (size_t)gridDim.x * blockDim.x;
    for (; i < n; i += stride) p[i] = 0.0f;
}

// ---------------------------------------------------------------------------
// In-degree: deg[dst] += 1.0f per edge
// ---------------------------------------------------------------------------
__global__ void deg_kernel(const int* __restrict__ dst, float* __restrict__ deg, int E) {
    int e = blockIdx.x * blockDim.x + threadIdx.x;
    if (e >= E) return;
    (void)__hip_atomic_fetch_add(&deg[dst[e]], 1.0f,
                                 __ATOMIC_RELAXED, __HIP_MEMORY_SCOPE_AGENT);
}

// ---------------------------------------------------------------------------
// Edge aggregation: S[dst] += h[src] - rel[etype]
// One wave (32 lanes) per edge; each lane handles a float4 (32*4 = 128 dims).
// Gathers hit a 25.6MB table (L2-resident on MI455X's 192MB L2).
// ---------------------------------------------------------------------------
__global__ __launch_bounds__(256)
void agg_kernel(const float* __restrict__ hin, const float* __restrict__ rel,
                const int* __restrict__ src, const int* __restrict__ dst,
                const int* __restrict__ etype, float* __restrict__ S, int E) {
    int gid  = blockIdx.x * blockDim.x + threadIdx.x;
    int e    = gid >> 5;          // one wave32 per edge
    int lane = gid & 31;
    if (e >= E) return;

    int s  = src[e];
    int d  = dst[e];
    int et = etype[e];

    const float4 hv = *(const float4*)(hin + (size_t)s  * H_DIM + lane * 4);
    const float4 rv = *(const float4*)(rel + (size_t)et * H_DIM + lane * 4);

    float* sp = S + (size_t)d * H_DIM + lane * 4;
    (void)__hip_atomic_fetch_add(&sp[0], hv.x - rv.x, __ATOMIC_RELAXED, __HIP_MEMORY_SCOPE_AGENT);
    (void)__hip_atomic_fetch_add(&sp[1], hv.y - rv.y, __ATOMIC_RELAXED, __HIP_MEMORY_SCOPE_AGENT);
    (void)__hip_atomic_fetch_add(&sp[2], hv.z - rv.z, __ATOMIC_RELAXED, __HIP_MEMORY_SCOPE_AGENT);
    (void)__hip_atomic_fetch_add(&sp[3], hv.w - rv.w, __ATOMIC_RELAXED, __HIP_MEMORY_SCOPE_AGENT);
}

// ---------------------------------------------------------------------------
// Fused RGCN layer GEMM (WMMA f32 16x16x4) + rrelu:
//   Hout[v] = rrelu( norm[v]*(S[v] @ Wn) + mask*h[v] @ Wl + (1-mask)*h[v] @ We )
// Block = 256 threads = 8 waves; block covers 16 rows x 128 cols,
// wave w computes the 16x16 tile at columns [16w, 16w+16).
// Fragment layouts per CDNA5 ISA 7.12.2 (wave32, f32):
//   A 16x4 : a.x = A[lid][k+2*half], a.y = A[lid][k+2*half+1]
//   B 4x16 : b.x = B[k+2*half][lid], b.y = B[k+2*half+1][lid]
//   C 16x16: c[r] -> row (r + 8*half), col lid
// ---------------------------------------------------------------------------
__global__ __launch_bounds__(256)
void rgcn_gemm_kernel(const float* __restrict__ S, const float* __restrict__ Hin,
                      const float* __restrict__ deg,
                      const float* __restrict__ Wn, const float* __restrict__ Wl,
                      const float* __restrict__ We, float* __restrict__ Hout) {
    const int m0   = blockIdx.x << 4;      // 16 rows per block
    const int wave = threadIdx.x >> 5;     // 0..7
    const int lane = threadIdx.x & 31;
    const int half = lane >> 4;            // 0 or 1
    const int lid  = lane & 15;
    const int n0   = wave << 4;            // 16 output cols per wave
    const int row  = m0 + lid;             // A-row this lane feeds

    const float dg   = deg[row];
    const float nrm  = 1.0f / fmaxf(dg, 1.0f);   // 1/in-degree
    const float mIn  = dg > 0.0f ? 1.0f : 0.0f;  // has_in mask
    const float mOut = 1.0f - mIn;

    const float* __restrict__ aS = S   + (size_t)row * H_DIM + 2 * half;
    const float* __restrict__ aH = Hin + (size_t)row * H_DIM + 2 * half;

    v8f c = {};

    // ---- source 0: (S * norm) @ W_neigh -------------------------------
    #pragma unroll
    for (int k = 0; k < H_DIM; k += 4) {
        const float2 av = *(const float2*)(aS + k);
        v2f a; a.x = av.x * nrm; a.y = av.y * nrm;
        v2f b;
        b.x = Wn[(size_t)(k + 2 * half)     * H_DIM + n0 + lid];
        b.y = Wn[(size_t)(k + 2 * half + 1) * H_DIM + n0 + lid];
        c = __builtin_amdgcn_wmma_f32_16x16x4_f32(false, a, false, b,
                                                  (short)0, c, false, false);
    }

    // ---- sources 1+2: mask*h @ loop_w  +  (1-mask)*h @ evolve_w -------
    #pragma unroll
    for (int k = 0; k < H_DIM; k += 4) {
        const float2 hv = *(const float2*)(aH + k);
        v2f a1; a1.x = hv.x * mIn;  a1.y = hv.y * mIn;
        v2f a2; a2.x = hv.x * mOut; a2.y = hv.y * mOut;
        v2f b1, b2;
        const size_t r0 = (size_t)(k + 2 * half) * H_DIM + n0 + lid;
        b1.x = Wl[r0]; b1.y = Wl[r0 + H_DIM];
        b2.x = We[r0]; b2.y = We[r0 + H_DIM];
        c = __builtin_amdgcn_wmma_f32_16x16x4_f32(false, a1, false, b1,
                                                  (short)0, c, false, false);
        c = __builtin_amdgcn_wmma_f32_16x16x4_f32(false, a2, false, b2,
                                                  (short)0, c, false, false);
    }

    // ---- rrelu epilogue + store ---------------------------------------
    #pragma unroll
    for (int r = 0; r < 8; ++r) {
        float v = c[r];
        v = (v >= 0.0f) ? v : v * RRELU_SLP;
        Hout[(size_t)(m0 + r + 8 * half) * H_DIM + n0 + lid] = v;
    }
}

// ---------------------------------------------------------------------------
// Final gather + time embedding:
//   out[pos][0:128]   = (id < N) ? h2[id] : 0      (id == N -> zero pad row)
//   out[pos][128:192] = cos(t * t_w + t_b)
// ---------------------------------------------------------------------------
__global__ __launch_bounds__(256)
void gather_kernel(const float* __restrict__ h2, const int* __restrict__ ids,
                   const int* __restrict__ times, const float* __restrict__ tw,
                   const float* __restrict__ tb, float* __restrict__ out,
                   int nPos, int nNodes) {
    int idx = blockIdx.x * blockDim.x + threadIdx.x;
    int total = nPos * 192;
    if (idx >= total) return;
    int pos = idx / 192;
    int j   = idx - pos * 192;
    float v;
    if (j < H_DIM) {
        int id = ids[pos];
        v = (id < nNodes) ? h2[(size_t)id * H_DIM + j] : 0.0f;
    } else {
        int jj = j - H_DIM;
        float t = (float)times[pos];
        v = cosf(t * tw[jj] + tb[jj]);
    }
    out[idx] = v;
}

// ---------------------------------------------------------------------------
// Host launcher
// ---------------------------------------------------------------------------
extern "C" void kernel_launch(void* const* d_in, const int* in_sizes, int n_in,
                              void* d_out, int out_size, void* d_ws, size_t ws_size,
                              hipStream_t stream) {
    const float* h     = (const float*)d_in[0];
    const float* rel   = (const float*)d_in[1];
    const float* wn1   = (const float*)d_in[2];
    const float* wl1   = (const float*)d_in[3];
    const float* we1   = (const float*)d_in[4];
    const float* wn2   = (const float*)d_in[5];
    const float* wl2   = (const float*)d_in[6];
    const float* we2   = (const float*)d_in[7];
    const float* tw    = (const float*)d_in[8];
    const float* tb    = (const float*)d_in[9];
    const int*   src   = (const int*)d_in[10];
    const int*   dst   = (const int*)d_in[11];
    const int*   etype = (const int*)d_in[12];
    const int*   ids   = (const int*)d_in[13];
    const int*   times = (const int*)d_in[14];
    float*       out   = (float*)d_out;

    const int N = in_sizes[0] / H_DIM;     // 50000 (divisible by 16)
    const int E = in_sizes[10];            // 800000
    const int P = in_sizes[13];            // BATCH*SEQ_LEN = 10240

    // workspace layout: deg[N] | S[N*128] | h1[N*128] | h2[N*128]
    float* deg = (float*)d_ws;
    float* S   = deg + N;
    float* h1  = S   + (size_t)N * H_DIM;
    float* h2  = h1  + (size_t)N * H_DIM;

    const int mTiles  = N / 16;                    // 3125
    const int aggGrid = (E * 32 + 255) / 256;      // wave per edge

    // degree (once)
    zero_f32<<<2048, 256, 0, stream>>>(deg, (size_t)N * (H_DIM + 1)); // deg + S
    deg_kernel<<<(E + 255) / 256, 256, 0, stream>>>(dst, deg, E);

    // ---- layer 1 ----
    agg_kernel<<<aggGrid, 256, 0, stream>>>(h, rel, src, dst, etype, S, E);
    rgcn_gemm_kernel<<<mTiles, 256, 0, stream>>>(S, h, deg, wn1, wl1, we1, h1);

    // ---- layer 2 ----
    zero_f32<<<2048, 256, 0, stream>>>(S, (size_t)N * H_DIM);
    agg_kernel<<<aggGrid, 256, 0, stream>>>(h1, rel, src, dst, etype, S, E);
    rgcn_gemm_kernel<<<mTiles, 256, 0, stream>>>(S, h1, deg, wn2, wl2, we2, h2);

    // ---- gather + time embedding ----
    gather_kernel<<<(P * 192 + 255) / 256, 256, 0, stream>>>(h2, ids, times, tw, tb,
                                                             out, P, N);
}